// GAT_59030030516771
// MI455X (gfx1250) — compile-verified
//
#include <hip/hip_runtime.h>
#include <math.h>

#define NEG_SLOPE 0.2f

typedef __attribute__((ext_vector_type(2))) float v2f;
typedef __attribute__((ext_vector_type(8))) float v8f;

// ---------------------------------------------------------------------------
// GEMM C[M,Ncols] = A[M,K] * B[K,Ncols] (+bias, optional ReLU) via
// V_WMMA_F32_16X16X4_F32. One wave32 per 16x16 output tile, K==64 fixed.
// A layout per ISA: lanes 0-15 -> M=lane, K={k0,k0+1}; lanes 16-31 -> M=lane-16, K={k0+2,k0+3}
// B layout symmetric over N; C/D: lane n=lane&15, VGPR r -> M = r + 8*(lane>>4)
// ---------------------------------------------------------------------------
template <bool RELU>
__global__ __launch_bounds__(256) void gemm_wmma_f32(const float* __restrict__ A,
                                                     const float* __restrict__ B,
                                                     const float* __restrict__ bias,
                                                     float* __restrict__ C,
                                                     int M, int Ncols, int K) {
  const int wave = (int)((blockIdx.x * blockDim.x + threadIdx.x) >> 5);
  const int lane = (int)(threadIdx.x & 31);
  const int tilesN = Ncols >> 4;
  const int trow = wave / tilesN;
  const int tcol = wave - trow * tilesN;
  if (trow * 16 >= M) return;  // wave-uniform exit: EXEC stays all-ones for WMMA

  const int half = lane >> 4;   // 0 or 1
  const int m    = lane & 15;
  const int arow = trow * 16 + m;   // A row this lane feeds
  const int bcol = tcol * 16 + m;   // B/C column this lane feeds

  v8f acc = {};
#pragma unroll
  for (int k0 = 0; k0 < 64; k0 += 4) {
    const int ka = k0 + 2 * half;
    v2f a, b;
    a.x = A[arow * K + ka];
    a.y = A[arow * K + ka + 1];
    b.x = B[ka * Ncols + bcol];
    b.y = B[(ka + 1) * Ncols + bcol];
    acc = __builtin_amdgcn_wmma_f32_16x16x4_f32(false, a, false, b, (short)0, acc,
                                                false, false);
  }

  const float bb = bias ? bias[bcol] : 0.0f;
#pragma unroll
  for (int r = 0; r < 8; ++r) {
    const int crow = trow * 16 + r + 8 * half;
    float v = acc[r] + bb;
    if (RELU) v = fmaxf(v, 0.0f);
    C[crow * Ncols + bcol] = v;
  }
}

// alpha_src[i] = h[i] . a_src ; alpha_dst[i] = h[i] . a_dst
__global__ __launch_bounds__(256) void attn_coef(const float* __restrict__ h,
                                                 const float* __restrict__ a_s,
                                                 const float* __restrict__ a_d,
                                                 float* __restrict__ as_out,
                                                 float* __restrict__ ad_out, int n) {
  const int i = blockIdx.x * blockDim.x + threadIdx.x;
  if (i >= n) return;
  const float* hp = h + (size_t)i * 64;
  float s = 0.f, d = 0.f;
#pragma unroll
  for (int f = 0; f < 64; ++f) {
    const float v = hp[f];
    s += v * a_s[f];
    d += v * a_d[f];
  }
  as_out[i] = s;
  ad_out[i] = d;
}

__global__ __launch_bounds__(256) void fill_f32(float* __restrict__ p, float v, int n) {
  const int i = blockIdx.x * blockDim.x + threadIdx.x;
  if (i < n) p[i] = v;
}

// pass 1: e = leaky_relu(as[src]+ad[dst]); segment-max into emax[dst]
__global__ __launch_bounds__(256) void edge_logits(const int* __restrict__ srci,
                                                   const int* __restrict__ dsti,
                                                   const float* __restrict__ as,
                                                   const float* __restrict__ ad,
                                                   float* __restrict__ e,
                                                   float* __restrict__ emax,
                                                   int Eorig, int Etot) {
  const int i = blockIdx.x * blockDim.x + threadIdx.x;
  if (i >= Etot) return;
  int s, d;
  if (i < Eorig) { s = srci[i]; d = dsti[i]; } else { s = d = i - Eorig; }
  float v = as[s] + ad[d];
  v = (v > 0.f) ? v : NEG_SLOPE * v;
  e[i] = v;
  // sign-aware float atomic max (monotone int repr for >=0, reversed uint for <0)
  if (v >= 0.f) atomicMax((int*)(emax + d), __float_as_int(v));
  else          atomicMin((unsigned int*)(emax + d), __float_as_uint(v));
}

// pass 2: e <- exp(e - emax[dst]); denom[dst] += e
__global__ __launch_bounds__(256) void edge_exp(const int* __restrict__ dsti,
                                                float* __restrict__ e,
                                                const float* __restrict__ emax,
                                                float* __restrict__ denom,
                                                int Eorig, int Etot) {
  const int i = blockIdx.x * blockDim.x + threadIdx.x;
  if (i >= Etot) return;
  const int d = (i < Eorig) ? dsti[i] : (i - Eorig);
  const float ee = expf(e[i] - emax[d]);
  e[i] = ee;
  atomicAdd(denom + d, ee);
}

// pass 3: out[dst] += (e/denom[dst]) * h[src]; one wave32 per edge, 2 feats/lane
__global__ __launch_bounds__(256) void edge_aggregate(const int* __restrict__ srci,
                                                      const int* __restrict__ dsti,
                                                      const float* __restrict__ e,
                                                      const float* __restrict__ denom,
                                                      const float* __restrict__ h,
                                                      float* __restrict__ out,
                                                      int Eorig, int Etot) {
  const long long gid = (long long)blockIdx.x * blockDim.x + threadIdx.x;
  const int edge = (int)(gid >> 5);
  const int lane = (int)(gid & 31);
  if (edge >= Etot) return;
  int s, d;
  if (edge < Eorig) { s = srci[edge]; d = dsti[edge]; } else { s = d = edge - Eorig; }
  const float alpha = e[edge] / (denom[d] + 1e-16f);
  const float v0 = alpha * h[(size_t)s * 64 + lane];
  const float v1 = alpha * h[(size_t)s * 64 + lane + 32];
  atomicAdd(out + (size_t)d * 64 + lane, v0);
  atomicAdd(out + (size_t)d * 64 + lane + 32, v1);
}

// out = relu(l2_normalize(in + bias)); one wave32 per node
__global__ __launch_bounds__(256) void bias_l2norm_relu(const float* __restrict__ in,
                                                        const float* __restrict__ bias,
                                                        float* __restrict__ outp, int n) {
  const long long gid = (long long)blockIdx.x * blockDim.x + threadIdx.x;
  const int node = (int)(gid >> 5);
  const int lane = (int)(gid & 31);
  if (node >= n) return;
  float v0 = in[(size_t)node * 64 + lane] + bias[lane];
  float v1 = in[(size_t)node * 64 + lane + 32] + bias[lane + 32];
  float ss = v0 * v0 + v1 * v1;
#pragma unroll
  for (int m = 16; m > 0; m >>= 1) ss += __shfl_xor(ss, m, 32);
  const float inv = 1.0f / fmaxf(sqrtf(ss), 1e-12f);
  outp[(size_t)node * 64 + lane]      = fmaxf(v0 * inv, 0.0f);
  outp[(size_t)node * 64 + lane + 32] = fmaxf(v1 * inv, 0.0f);
}

// global add pool: g[batch[i]] += h[i]; one wave32 per node
__global__ __launch_bounds__(256) void pool_sum(const float* __restrict__ h,
                                                const int* __restrict__ batch,
                                                float* __restrict__ g, int n) {
  const long long gid = (long long)blockIdx.x * blockDim.x + threadIdx.x;
  const int node = (int)(gid >> 5);
  const int lane = (int)(gid & 31);
  if (node >= n) return;
  const int b = batch[node];
  atomicAdd(g + (size_t)b * 64 + lane,      h[(size_t)node * 64 + lane]);
  atomicAdd(g + (size_t)b * 64 + lane + 32, h[(size_t)node * 64 + lane + 32]);
}

// logits = g @ fc2_w + fc2_b; out = log_softmax(logits). One thread per graph.
__global__ __launch_bounds__(256) void fc2_logsoftmax(const float* __restrict__ g,
                                                      const float* __restrict__ w,
                                                      const float* __restrict__ b,
                                                      float* __restrict__ out,
                                                      int nG, int nC) {
  const int i = blockIdx.x * blockDim.x + threadIdx.x;
  if (i >= nG) return;
  float logit[10];
#pragma unroll
  for (int c = 0; c < 10; ++c) logit[c] = b[c];
  for (int f = 0; f < 64; ++f) {
    const float gv = g[(size_t)i * 64 + f];
#pragma unroll
    for (int c = 0; c < 10; ++c) logit[c] += gv * w[f * 10 + c];
  }
  float mx = logit[0];
#pragma unroll
  for (int c = 1; c < 10; ++c) mx = fmaxf(mx, logit[c]);
  float sum = 0.f;
#pragma unroll
  for (int c = 0; c < 10; ++c) sum += expf(logit[c] - mx);
  const float lse = mx + logf(sum);
#pragma unroll
  for (int c = 0; c < 10; ++c) out[(size_t)i * 10 + c] = logit[c] - lse;
}

extern "C" void kernel_launch(void* const* d_in, const int* in_sizes, int n_in,
                              void* d_out, int out_size, void* d_ws, size_t ws_size,
                              hipStream_t stream) {
  const float* x     = (const float*)d_in[0];
  const int*   ei    = (const int*)d_in[1];   // [2, E] int32
  const int*   batch = (const int*)d_in[2];
  const float* w[3]   = {(const float*)d_in[3],  (const float*)d_in[7],  (const float*)d_in[11]};
  const float* asv[3] = {(const float*)d_in[4],  (const float*)d_in[8],  (const float*)d_in[12]};
  const float* adv[3] = {(const float*)d_in[5],  (const float*)d_in[9],  (const float*)d_in[13]};
  const float* bv[3]  = {(const float*)d_in[6],  (const float*)d_in[10], (const float*)d_in[14]};
  const float* fc1_w = (const float*)d_in[15];
  const float* fc1_b = (const float*)d_in[16];
  const float* fc2_w = (const float*)d_in[17];
  const float* fc2_b = (const float*)d_in[18];

  const int N   = in_sizes[0] / 64;
  const int E   = in_sizes[1] / 2;
  const int ET  = E + N;        // edges + self loops
  const int GG  = 256, DIM = 64, CC = 10;

  // workspace carve-out (256B aligned slices)
  char* ws = (char*)d_ws;
  auto carve = [&](size_t bytes) {
    char* p = ws;
    ws += (bytes + 255) & ~(size_t)255;
    return p;
  };
  float* bufA  = (float*)carve((size_t)N * DIM * 4);   // transformed features h
  float* bufB  = (float*)carve((size_t)N * DIM * 4);   // aggregation accumulator
  float* asb   = (float*)carve((size_t)N * 4);
  float* adb   = (float*)carve((size_t)N * 4);
  float* ebuf  = (float*)carve((size_t)ET * 4);
  float* emax  = (float*)carve((size_t)N * 4);
  float* denom = (float*)carve((size_t)N * 4);
  float* gpool = (float*)carve((size_t)GG * DIM * 4);
  float* fc1o  = (float*)carve((size_t)GG * DIM * 4);

  const int* srci = ei;
  const int* dsti = ei + E;

  const int TB = 256;
  const int tilesRow = (N + 15) / 16;
  const int gemmWaves = tilesRow * (DIM / 16);
  const int gemmBlocks = (gemmWaves * 32 + TB - 1) / TB;
  const int nodeBlocks  = (N + TB - 1) / TB;
  const int edgeBlocks  = (ET + TB - 1) / TB;
  const int nodeWaveBlk = (int)(((long long)N * 32 + TB - 1) / TB);
  const int edgeWaveBlk = (int)(((long long)ET * 32 + TB - 1) / TB);
  const int featBlocks  = (int)(((long long)N * DIM + TB - 1) / TB);

  const float* cur = x;
  for (int L = 0; L < 3; ++L) {
    gemm_wmma_f32<false><<<gemmBlocks, TB, 0, stream>>>(cur, w[L], nullptr, bufA, N, DIM, DIM);
    attn_coef<<<nodeBlocks, TB, 0, stream>>>(bufA, asv[L], adv[L], asb, adb, N);
    fill_f32<<<nodeBlocks, TB, 0, stream>>>(emax, -3.0e38f, N);
    fill_f32<<<nodeBlocks, TB, 0, stream>>>(denom, 0.0f, N);
    fill_f32<<<featBlocks, TB, 0, stream>>>(bufB, 0.0f, N * DIM);
    edge_logits<<<edgeBlocks, TB, 0, stream>>>(srci, dsti, asb, adb, ebuf, emax, E, ET);
    edge_exp<<<edgeBlocks, TB, 0, stream>>>(dsti, ebuf, emax, denom, E, ET);
    edge_aggregate<<<edgeWaveBlk, TB, 0, stream>>>(srci, dsti, ebuf, denom, bufA, bufB, E, ET);
    bias_l2norm_relu<<<nodeWaveBlk, TB, 0, stream>>>(bufB, bv[L], bufA, N);
    cur = bufA;
  }

  fill_f32<<<(GG * DIM + TB - 1) / TB, TB, 0, stream>>>(gpool, 0.0f, GG * DIM);
  pool_sum<<<nodeWaveBlk, TB, 0, stream>>>(bufA, batch, gpool, N);

  const int fcWaves = (GG / 16) * (DIM / 16);
  gemm_wmma_f32<true><<<(fcWaves * 32 + TB - 1) / TB, TB, 0, stream>>>(gpool, fc1_w, fc1_b,
                                                                       fc1o, GG, DIM, DIM);
  fc2_logsoftmax<<<(GG + TB - 1) / TB, TB, 0, stream>>>(fc1o, fc2_w, fc2_b, (float*)d_out,
                                                        GG, CC);
}